// MultiMarginRankingLoss_24266565222505
// MI455X (gfx1250) — compile-verified
//
#include <hip/hip_runtime.h>
#include <hip/hip_bf16.h>
#include <stdint.h>

#define NUM_Q 8192
#define NUM_P 4096
#define TPB   256   // 8 wave32s

__device__ __forceinline__ float wave_min32(float v) {
#pragma unroll
    for (int m = 16; m > 0; m >>= 1)
        v = fminf(v, __shfl_xor(v, m, 32));
    return v;
}

__device__ __forceinline__ float wave_sum32(float v) {
#pragma unroll
    for (int m = 16; m > 0; m >>= 1)
        v += __shfl_xor(v, m, 32);
    return v;
}

// One block per row. Async-DMA the pred row into LDS (read it twice from
// there), overlap y loads, then min-reduce over relevant and hinge-sum over
// non-relevant.
__global__ __launch_bounds__(TPB) void row_hinge_kernel(
        const float* __restrict__ pred,
        const int*   __restrict__ y,
        float*       __restrict__ partial) {
    __shared__ __align__(16) float spred[NUM_P];   // 16 KB staged pred row
    __shared__ float smin[TPB / 32];
    __shared__ float ssum[TPB / 32];

    const int row  = blockIdx.x;
    const int t    = threadIdx.x;
    const int wave = t >> 5;
    const int lane = t & 31;

    const float* prow = pred + (size_t)row * NUM_P;
    const int*   yrow = y    + (size_t)row * NUM_P;

    // Low 32 bits of the flat shared address == workgroup-relative LDS offset.
    const unsigned lds_base = (unsigned)(uintptr_t)(&spred[0]);

    // Issue async global->LDS DMA: 4 x b128 per thread = 4096 floats total.
    // Tracked by ASYNCcnt; no VGPR data round-trip.
#pragma unroll
    for (int c = 0; c < 4; ++c) {
        const int      v4     = c * TPB + t;                    // float4 index
        const unsigned ldsoff = lds_base + (unsigned)(v4 * 16); // bytes
        const float*   g      = prow + v4 * 4;
        asm volatile("global_load_async_to_lds_b128 %0, %1, off"
                     :: "v"(ldsoff), "v"(g) : "memory");
    }

    // Overlap: load y (vectorized) while the DMA is in flight; compress the
    // relevance of this thread's 16 elements into a 16-bit mask.
    unsigned mask = 0u;
#pragma unroll
    for (int c = 0; c < 4; ++c) {
        const int4 yv = ((const int4*)yrow)[c * TPB + t];
        mask |= (yv.x == 1 ? 1u : 0u) << (c * 4 + 0);
        mask |= (yv.y == 1 ? 1u : 0u) << (c * 4 + 1);
        mask |= (yv.z == 1 ? 1u : 0u) << (c * 4 + 2);
        mask |= (yv.w == 1 ? 1u : 0u) << (c * 4 + 3);
    }

    // Wait only for the async DMA (split counter), then make LDS visible
    // workgroup-wide.
    asm volatile("s_wait_asynccnt 0" ::: "memory");
    __syncthreads();

    // Pass A: min of pred over relevant entries (every row has >= 1).
    float lmin = __builtin_inff();
#pragma unroll
    for (int c = 0; c < 4; ++c) {
        const float4 p = ((const float4*)spred)[c * TPB + t];
        if (mask & (1u << (c * 4 + 0))) lmin = fminf(lmin, p.x);
        if (mask & (1u << (c * 4 + 1))) lmin = fminf(lmin, p.y);
        if (mask & (1u << (c * 4 + 2))) lmin = fminf(lmin, p.z);
        if (mask & (1u << (c * 4 + 3))) lmin = fminf(lmin, p.w);
    }
    lmin = wave_min32(lmin);
    if (lane == 0) smin[wave] = lmin;
    __syncthreads();

    float rowmin = smin[0];
#pragma unroll
    for (int w = 1; w < TPB / 32; ++w) rowmin = fminf(rowmin, smin[w]);

    // Pass B: hinge sum over non-relevant entries, re-read pred from LDS.
    float lsum = 0.0f;
#pragma unroll
    for (int c = 0; c < 4; ++c) {
        const float4 p = ((const float4*)spred)[c * TPB + t];
        if (!(mask & (1u << (c * 4 + 0)))) lsum += fmaxf(p.x - rowmin, 0.0f);
        if (!(mask & (1u << (c * 4 + 1)))) lsum += fmaxf(p.y - rowmin, 0.0f);
        if (!(mask & (1u << (c * 4 + 2)))) lsum += fmaxf(p.z - rowmin, 0.0f);
        if (!(mask & (1u << (c * 4 + 3)))) lsum += fmaxf(p.w - rowmin, 0.0f);
    }
    lsum = wave_sum32(lsum);
    if (lane == 0) ssum[wave] = lsum;
    __syncthreads();

    if (t == 0) {
        float s = 0.0f;
#pragma unroll
        for (int w = 0; w < TPB / 32; ++w) s += ssum[w];
        partial[row] = s;   // raw hinge sum for this row
    }
}

// Deterministic fixed-order reduction of the 8192 per-row sums.
__global__ __launch_bounds__(TPB) void final_reduce_kernel(
        const float* __restrict__ partial,
        float*       __restrict__ out) {
    __shared__ float s[TPB];
    float acc = 0.0f;
    for (int i = threadIdx.x; i < NUM_Q; i += TPB)
        acc += partial[i];
    s[threadIdx.x] = acc;
    __syncthreads();
#pragma unroll
    for (int off = TPB / 2; off > 0; off >>= 1) {
        if (threadIdx.x < off) s[threadIdx.x] += s[threadIdx.x + off];
        __syncthreads();
    }
    if (threadIdx.x == 0)
        out[0] = s[0] * (1.0f / ((float)NUM_Q * (float)NUM_P));
}

extern "C" void kernel_launch(void* const* d_in, const int* in_sizes, int n_in,
                              void* d_out, int out_size, void* d_ws, size_t ws_size,
                              hipStream_t stream) {
    const float* pred = (const float*)d_in[0];
    const int*   y    = (const int*)d_in[1];
    float* out        = (float*)d_out;
    float* partial    = (float*)d_ws;   // NUM_Q floats = 32 KB scratch

    row_hinge_kernel<<<NUM_Q, TPB, 0, stream>>>(pred, y, partial);
    final_reduce_kernel<<<1, TPB, 0, stream>>>(partial, out);
}